// CongestionWrapperEncoder1_40089224741035
// MI455X (gfx1250) — compile-verified
//
#include <hip/hip_runtime.h>
#include <hip/hip_bf16.h>

// ---------------- problem constants (from reference) ----------------
#define Bc   4
#define Dc   4
#define Gc   (Bc * Dc)   // 16 graphs
#define Fc   64          // in_channels
#define Cc   32          // out channels per head
#define Hh   4           // heads
#define HC   (Hh * Cc)   // 128
#define NEG_SLOPE 0.2f

typedef __attribute__((ext_vector_type(2))) float v2f;
typedef __attribute__((ext_vector_type(8))) float v8f;

// ---------------- helpers ----------------
__device__ __forceinline__ float leaky(float v) {
    return v > 0.0f ? v : NEG_SLOPE * v;
}
// monotonic float -> uint key (total order incl. negatives) for atomicMax
__device__ __forceinline__ unsigned fkey(float f) {
    unsigned u = __float_as_uint(f);
    return (u & 0x80000000u) ? ~u : (u | 0x80000000u);
}
__device__ __forceinline__ float funkey(unsigned k) {
    unsigned u = (k & 0x80000000u) ? (k & 0x7FFFFFFFu) : ~k;
    return __uint_as_float(u);
}

// ---------------- 1) embW = emb(V x F) @ W(F x HC) via fp32 WMMA ----------------
// One wave per 16x16 output tile.  A: 16x4 f32 frag (lanes 0-15: K=0,1 ; lanes
// 16-31: K=2,3).  B: 4x16 frag (lane = column, halves select K pair).
__global__ void gemm_embW(const float* __restrict__ emb, const float* __restrict__ W,
                          float* __restrict__ embW) {
    const int colTiles = HC / 16;                 // 8
    int tile = blockIdx.x;                        // (V/16)*colTiles waves total
    int mt = tile / colTiles;
    int nt = tile % colTiles;
    int lane = threadIdx.x & 31;
    int half = lane >> 4;                         // 0 or 1
    int l16  = lane & 15;
    int m0 = mt * 16, n0 = nt * 16;

    v8f acc = {};
#pragma unroll
    for (int k0 = 0; k0 < Fc; k0 += 4) {
        int ka = k0 + half * 2;                   // this half-wave's K pair
        v2f a, b;
        a.x = emb[(m0 + l16) * Fc + ka + 0];      // A[m][ka], A[m][ka+1]
        a.y = emb[(m0 + l16) * Fc + ka + 1];
        b.x = W[(ka + 0) * HC + n0 + l16];        // B[ka][n], B[ka+1][n]
        b.y = W[(ka + 1) * HC + n0 + l16];
        acc = __builtin_amdgcn_wmma_f32_16x16x4_f32(
                  false, a, false, b, (short)0, acc, false, false);
    }
    // C/D layout: VGPR r -> M = r (lanes 0-15) / M = 8+r (lanes 16-31), N = l16
#pragma unroll
    for (int r = 0; r < 8; ++r) {
        int m = m0 + r + half * 8;
        embW[m * HC + n0 + l16] = acc[r];
    }
}

// ---------------- 2) aS[v,h] = embW[v,h,:]·att_src[h,:]  (same for aD) ----------
__global__ void att_proj(const float* __restrict__ embW,
                         const float* __restrict__ att_src,
                         const float* __restrict__ att_dst,
                         float* __restrict__ aS, float* __restrict__ aD, int VH) {
    int idx = blockIdx.x * blockDim.x + threadIdx.x;   // v*H + h
    if (idx >= VH) return;
    int v = idx / Hh, h = idx % Hh;
    float s = 0.f, d = 0.f;
#pragma unroll
    for (int c = 0; c < Cc; ++c) {
        float xv = embW[v * HC + h * Cc + c];
        s += xv * att_src[h * Cc + c];
        d += xv * att_dst[h * Cc + c];
    }
    aS[idx] = s;
    aD[idx] = d;
}

// ---------------- 3) init output (bias) and softmax stats ----------------
__global__ void init_out(float* __restrict__ out, const float* __restrict__ bias,
                         long long total) {
    long long i = (long long)blockIdx.x * blockDim.x + threadIdx.x;
    if (i < total) out[i] = bias[(int)(i & (HC - 1))];
}
__global__ void init_stats(unsigned* __restrict__ maxK, float* __restrict__ denom,
                           int total) {
    int i = blockIdx.x * blockDim.x + threadIdx.x;
    if (i < total) { maxK[i] = 0u; denom[i] = 0.f; }
}

// ---------------- 4) segment max over destinations ----------------
__global__ void edge_max(const int* __restrict__ x, const int* __restrict__ adj,
                         const float* __restrict__ aS, const float* __restrict__ aD,
                         unsigned* __restrict__ maxK, int N, int E, int Et) {
    long long gid = (long long)blockIdx.x * blockDim.x + threadIdx.x;
    long long total = (long long)Gc * Et;
    if (gid >= total) return;
    int g = (int)(gid / Et);
    int e = (int)(gid % Et);
    int s, d;
    if (e < E) { s = adj[e]; d = adj[E + e]; } else { s = d = e - E; }  // self loop
    int cs = x[g * N + s];
    int cd = x[g * N + d];
    unsigned* mrow = maxK + ((long long)g * N + d) * Hh;
#pragma unroll
    for (int h = 0; h < Hh; ++h) {
        float v = leaky(aS[cs * Hh + h] + aD[cd * Hh + h]);
        atomicMax(&mrow[h], fkey(v));
    }
}

// ---------------- 5) denom = sum exp(e - max) per destination ----------------
__global__ void edge_denom(const int* __restrict__ x, const int* __restrict__ adj,
                           const float* __restrict__ aS, const float* __restrict__ aD,
                           const unsigned* __restrict__ maxK, float* __restrict__ denom,
                           int N, int E, int Et) {
    long long gid = (long long)blockIdx.x * blockDim.x + threadIdx.x;
    long long total = (long long)Gc * Et;
    if (gid >= total) return;
    int g = (int)(gid / Et);
    int e = (int)(gid % Et);
    int s, d;
    if (e < E) { s = adj[e]; d = adj[E + e]; } else { s = d = e - E; }
    int cs = x[g * N + s];
    int cd = x[g * N + d];
    long long base = ((long long)g * N + d) * Hh;
#pragma unroll
    for (int h = 0; h < Hh; ++h) {
        float v = leaky(aS[cs * Hh + h] + aD[cd * Hh + h]);
        float m = funkey(maxK[base + h]);
        atomicAdd(&denom[base + h], expf(v - m));
    }
}

// ---------------- 6) message pass: wave32 per (g, edge), 128 channels --------
__global__ void edge_msg(const int* __restrict__ x, const int* __restrict__ adj,
                         const float* __restrict__ aS, const float* __restrict__ aD,
                         const unsigned* __restrict__ maxK, const float* __restrict__ denom,
                         const float* __restrict__ embW, float* __restrict__ out,
                         int N, int E, int Et) {
    int lane = threadIdx.x & 31;
    long long wid = ((long long)blockIdx.x * blockDim.x + threadIdx.x) >> 5;
    long long total = (long long)Gc * Et;
    if (wid >= total) return;
    int g = (int)(wid / Et);
    int e = (int)(wid % Et);
    int s, d;
    if (e < E) { s = adj[e]; d = adj[E + e]; } else { s = d = e - E; }
    int cs = x[g * N + s];
    int cd = x[g * N + d];
    long long base = ((long long)g * N + d) * Hh;
    float scale[Hh];
#pragma unroll
    for (int h = 0; h < Hh; ++h) {
        float v = leaky(aS[cs * Hh + h] + aD[cd * Hh + h]);
        float m = funkey(maxK[base + h]);
        scale[h] = expf(v - m) / denom[base + h];
    }
    const float* row  = embW + (long long)cs * HC;
    float*       orow = out  + ((long long)g * N + d) * HC;
#pragma unroll
    for (int i = 0; i < Hh; ++i) {                 // channel = i*32 + lane, head = i
        int ch = i * 32 + lane;
        atomicAdd(&orow[ch], scale[i] * row[ch]);
    }
}

// ---------------- host launch ----------------
extern "C" void kernel_launch(void* const* d_in, const int* in_sizes, int n_in,
                              void* d_out, int out_size, void* d_ws, size_t ws_size,
                              hipStream_t stream) {
    const int*   x       = (const int*)  d_in[0];   // [B,D,N] int32
    const int*   adj     = (const int*)  d_in[1];   // [2,E]   int32
    const float* emb     = (const float*)d_in[2];   // [V,F]
    const float* W       = (const float*)d_in[3];   // [F,H*C]
    const float* att_src = (const float*)d_in[4];   // [H,C]
    const float* att_dst = (const float*)d_in[5];   // [H,C]
    const float* bias    = (const float*)d_in[6];   // [H*C]
    float* out = (float*)d_out;

    const int N  = in_sizes[0] / Gc;        // 5000
    const int E  = in_sizes[1] / 2;         // 80000
    const int V  = in_sizes[2] / Fc;        // 256
    const int Et = E + N;                   // with self loops

    // workspace layout (floats)
    float*    embW  = (float*)d_ws;                       // V*HC
    float*    aS    = embW + (size_t)V * HC;              // V*Hh
    float*    aD    = aS   + (size_t)V * Hh;              // V*Hh
    unsigned* maxK  = (unsigned*)(aD + (size_t)V * Hh);   // Gc*N*Hh
    float*    denom = (float*)(maxK + (size_t)Gc * N * Hh);

    // 1) tiny fp32 WMMA GEMM: embW = emb @ W   (one wave per 16x16 tile)
    {
        int tiles = (V / 16) * (HC / 16);     // 128
        gemm_embW<<<tiles, 32, 0, stream>>>(emb, W, embW);
    }
    // 2) attention projections per congestion value
    {
        int VH = V * Hh;
        att_proj<<<(VH + 255) / 256, 256, 0, stream>>>(embW, att_src, att_dst, aS, aD, VH);
    }
    // 3) init output with bias; init softmax stats
    {
        long long total = (long long)Gc * N * HC;
        init_out<<<(unsigned)((total + 255) / 256), 256, 0, stream>>>(out, bias, total);
        int st = Gc * N * Hh;
        init_stats<<<(st + 255) / 256, 256, 0, stream>>>(maxK, denom, st);
    }
    // 4-5) softmax statistics over edges
    {
        long long total = (long long)Gc * Et;
        unsigned blocks = (unsigned)((total + 255) / 256);
        edge_max  <<<blocks, 256, 0, stream>>>(x, adj, aS, aD, maxK, N, E, Et);
        edge_denom<<<blocks, 256, 0, stream>>>(x, adj, aS, aD, maxK, denom, N, E, Et);
    }
    // 6) weighted scatter of messages: one wave per (graph, edge)
    {
        long long waves   = (long long)Gc * Et;
        long long threads = waves * 32;
        unsigned  blocks  = (unsigned)((threads + 255) / 256);
        edge_msg<<<blocks, 256, 0, stream>>>(x, adj, aS, aD, maxK, denom, embW, out, N, E, Et);
    }
}